// AdditiveAttention_38680475467955
// MI455X (gfx1250) — compile-verified
//
#include <hip/hip_runtime.h>
#include <hip/hip_bf16.h>

#define BB   8
#define LL   512
#define DK   64
#define KPAD 68   // 272B rows: 16B aligned, modest bank conflict

typedef float v2f __attribute__((ext_vector_type(2)));
typedef float v8f __attribute__((ext_vector_type(8)));

// Hardware tanh (CDNA5 V_TANH_F32, TRANS pipe).
__device__ __forceinline__ float htanh(float x) {
#if defined(__AMDGCN__) && __has_builtin(__builtin_amdgcn_tanhf)
  return __builtin_amdgcn_tanhf(x);
#else
  float r;
  // TRANS op: require 1 independent op / V_NOP before result use (ISA 7.4.3).
  asm volatile("v_tanh_f32 %0, %1\n\tv_nop" : "=v"(r) : "v"(x));
  return r;
#endif
}

// ---------------------------------------------------------------------------
// Kernel 1: q_proj / k_proj.  Y[row][e] = sum_d X[row][d] * W[e][d]
// blockIdx.y selects (Q,W_q)->qp or (K,W_k)->kp.  Tiny; W staged in LDS.
// ---------------------------------------------------------------------------
__global__ __launch_bounds__(256) void proj_kernel(
    const float* __restrict__ Q, const float* __restrict__ K,
    const float* __restrict__ Wq, const float* __restrict__ Wk,
    float* __restrict__ qp, float* __restrict__ kp) {
  __shared__ float Wsh[DK][DK + 1];   // +1 pad -> conflict-free
  __shared__ float Xsh[4][DK];
  const float* X; const float* W; float* Y;
  if (blockIdx.y == 0) { X = Q; W = Wq; Y = qp; } else { X = K; W = Wk; Y = kp; }
  const int tid = threadIdx.x;
  for (int i = tid; i < DK * DK; i += 256) Wsh[i >> 6][i & 63] = W[i];
  const int row0 = blockIdx.x * 4;
  for (int i = tid; i < 4 * DK; i += 256)
    Xsh[i >> 6][i & 63] = X[(size_t)(row0 + (i >> 6)) * DK + (i & 63)];
  __syncthreads();
  const int rl = tid >> 6;   // 0..3
  const int e  = tid & 63;
  float s = 0.f;
#pragma unroll
  for (int d = 0; d < DK; ++d) s = fmaf(Xsh[rl][d], Wsh[e][d], s);
  Y[(size_t)(row0 + rl) * DK + e] = s;
}

// ---------------------------------------------------------------------------
// Kernel 2: scores + softmax (the tanh-bound core).
// Grid (B, L/16); 512 threads; k_proj[b] (128KB) staged once per block in
// dynamic LDS (CDNA5: 320KB/WGP).  Thread k owns column k of each score row.
// ---------------------------------------------------------------------------
__global__ __launch_bounds__(512) void score_softmax_kernel(
    const float* __restrict__ qp, const float* __restrict__ kp,
    const float* __restrict__ vw, float* __restrict__ attn) {
  extern __shared__ float smem[];
  float* kpsh = smem;                    // [LL][KPAD]
  float* qpsh = smem + LL * KPAD;        // [64]
  float* vwsh = qpsh + DK;               // [64]
  float* red  = vwsh + DK;               // [16] wave partials

  const int b   = blockIdx.x;
  const int qt  = blockIdx.y;
  const int tid = threadIdx.x;           // 0..511 == k index
  const int lane = tid & 31, wid = tid >> 5;

  if (tid < DK) vwsh[tid] = vw[tid];
  const float4* g4 = (const float4*)(kp + (size_t)b * LL * DK);
  for (int i = tid; i < LL * (DK / 4); i += 512) {
    const int k = i >> 4, d4 = i & 15;
    *(float4*)&kpsh[k * KPAD + d4 * 4] = g4[i];
  }
  __syncthreads();

  const float* kr = &kpsh[tid * KPAD];
  for (int r = 0; r < 16; ++r) {
    const int qrow = qt * 16 + r;
    if (tid < DK) qpsh[tid] = qp[((size_t)b * LL + qrow) * DK + tid];
    __syncthreads();

    float s = 0.f;
#pragma unroll
    for (int d = 0; d < DK; d += 4) {
      const float4 kv = *(const float4*)&kr[d];
      const float4 qv = *(const float4*)&qpsh[d];   // LDS broadcast
      const float4 wv = *(const float4*)&vwsh[d];   // LDS broadcast
      s = fmaf(wv.x, htanh(qv.x + kv.x), s);
      s = fmaf(wv.y, htanh(qv.y + kv.y), s);
      s = fmaf(wv.z, htanh(qv.z + kv.z), s);
      s = fmaf(wv.w, htanh(qv.w + kv.w), s);
    }

    // --- softmax over 512 values (one per thread): wave32 shfl + LDS ---
    float m = s;
    for (int off = 16; off; off >>= 1) m = fmaxf(m, __shfl_xor(m, off, 32));
    if (lane == 0) red[wid] = m;
    __syncthreads();
    float m2 = red[0];
#pragma unroll
    for (int w = 1; w < 16; ++w) m2 = fmaxf(m2, red[w]);

    const float e = __expf(s - m2);
    float t = e;
    for (int off = 16; off; off >>= 1) t += __shfl_xor(t, off, 32);
    __syncthreads();
    if (lane == 0) red[wid] = t;
    __syncthreads();
    float sum = 0.f;
#pragma unroll
    for (int w = 0; w < 16; ++w) sum += red[w];

    attn[((size_t)b * LL + qrow) * LL + tid] = e / sum;
    __syncthreads();   // protect qpsh/red before next row
  }
}

// ---------------------------------------------------------------------------
// Kernel 3: out = attn @ V via V_WMMA_F32_16X16X4_F32.
// One 16x16 output tile per wave; K=512 in steps of 4 (128 WMMA/tile).
// A lane layout (ISA 7.12.2): A[m=lane%16][k0 + 2*(lane/16) + j], j=0,1.
// B: V[k0 + 2*(lane/16) + j][n=lane%16].  C/D: vgpr r -> M=r+8*(lane/16).
// ---------------------------------------------------------------------------
__global__ __launch_bounds__(128) void av_wmma_kernel(
    const float* __restrict__ attn, const float* __restrict__ V,
    float* __restrict__ out) {
  const int wave = threadIdx.x >> 5;
  const int lane = threadIdx.x & 31;
  const int half = lane >> 4;       // 0|1
  const int l16  = lane & 15;

  const int tile = blockIdx.x * 4 + wave;   // 0..1023
  const int b  = tile >> 7;                 // /128
  const int mt = (tile & 127) >> 2;         // 0..31
  const int nt = tile & 3;                  // 0..3

  const float* A  = attn + (size_t)b * LL * LL;   // 512x512
  const float* Bm = V    + (size_t)b * LL * DK;   // 512x64
  const int arow = mt * 16 + l16;
  const int ncol = nt * 16 + l16;

  v8f c = {0.f, 0.f, 0.f, 0.f, 0.f, 0.f, 0.f, 0.f};
#pragma unroll 4
  for (int k0 = 0; k0 < LL; k0 += 4) {
    const int ka = k0 + 2 * half;                      // even -> b64 aligned
    const v2f a = *(const v2f*)&A[(size_t)arow * LL + ka];
    v2f bf;
    bf.x = Bm[(size_t)(ka + 0) * DK + ncol];
    bf.y = Bm[(size_t)(ka + 1) * DK + ncol];
    c = __builtin_amdgcn_wmma_f32_16x16x4_f32(
        /*neg_a=*/false, a, /*neg_b=*/false, bf,
        /*c_mod=*/(short)0, c, /*reuse_a=*/false, /*reuse_b=*/false);
  }
#pragma unroll
  for (int r = 0; r < 8; ++r) {
    const int m = mt * 16 + r + 8 * half;
    out[((size_t)b * LL + m) * DK + ncol] = c[r];
  }
}

// ---------------------------------------------------------------------------
extern "C" void kernel_launch(void* const* d_in, const int* in_sizes, int n_in,
                              void* d_out, int out_size, void* d_ws, size_t ws_size,
                              hipStream_t stream) {
  const float* Q  = (const float*)d_in[0];
  const float* K  = (const float*)d_in[1];
  const float* V  = (const float*)d_in[2];
  const float* Wq = (const float*)d_in[3];
  const float* Wk = (const float*)d_in[4];
  const float* vw = (const float*)d_in[5];

  float* out  = (float*)d_out;                       // [8,512,64]
  float* attn = out + (size_t)BB * LL * DK;          // [8,512,512]

  float* qp = (float*)d_ws;                          // [8*512,64]
  float* kp = qp + (size_t)BB * LL * DK;             // [8*512,64]

  // 1) projections (both in one launch; blockIdx.y selects Q or K path)
  proj_kernel<<<dim3((BB * LL) / 4, 2), 256, 0, stream>>>(Q, K, Wq, Wk, qp, kp);

  // 2) tanh-scores + softmax; 139.5KB dynamic LDS per block (CDNA5 320KB/WGP)
  const size_t sm = (size_t)(LL * KPAD + DK + DK + 16) * sizeof(float);
  score_softmax_kernel<<<dim3(BB, LL / 16), 512, sm, stream>>>(qp, kp, vw, attn);

  // 3) out = attn @ V with f32 WMMA
  av_wmma_kernel<<<dim3(256), 128, 0, stream>>>(attn, V, out);
}